// AdapterLayer_25872882991396
// MI455X (gfx1250) — compile-verified
//
#include <hip/hip_runtime.h>
#include <hip/hip_bf16.h>

#define WIDTH 1024
#define NH 16
#define HD 64
#define FF 256          // 4*HD
#define MBS 16
#define BB 4
#define LB 2048
#define NCHUNK (LB/MBS) // 128
#define EPSF 1e-6f
#define BASE_LRF 1.0f

#if __has_builtin(__builtin_amdgcn_global_load_async_to_lds_b128) && \
    __has_builtin(__builtin_amdgcn_s_wait_asynccnt)
#define USE_ASYNC_LDS 1
#else
#define USE_ASYNC_LDS 0
#endif

typedef float v2f __attribute__((ext_vector_type(2)));
typedef float v8f __attribute__((ext_vector_type(8)));
typedef int   v4i __attribute__((ext_vector_type(4)));

#if USE_ASYNC_LDS
typedef __attribute__((address_space(1))) v4i* gas_v4i_ptr;  // global (AS1)
typedef __attribute__((address_space(3))) v4i* lds_v4i_ptr;  // LDS (AS3)
#endif

__device__ __forceinline__ v8f wmma4(v2f a, v2f b, v8f c) {
  // V_WMMA_F32_16X16X4_F32 : D(16x16,f32) = A(16x4,f32) x B(4x16,f32) + C
  return __builtin_amdgcn_wmma_f32_16x16x4_f32(false, a, false, b, (short)0, c, false, false);
}

// branch-free tanh on the hardware transcendental pipe (v_exp_f32 + v_rcp_f32).
// saturates correctly: x->+inf => 1, x->-inf => -1. ~1e-6 rel error.
__device__ __forceinline__ float tanh_fast(float x) {
  float e = __expf(2.f * x);
  return 1.f - 2.f * __builtin_amdgcn_rcpf(e + 1.f);
}
__device__ __forceinline__ float gelu_f(float x) {
  return 0.5f * x * (1.f + tanh_fast(0.7978845608f * (x + 0.044715f * x * x * x)));
}
__device__ __forceinline__ float gelu_bwd_f(float x) {
  float t = tanh_fast(0.79788456f * x * (1.f + 0.044715f * x * x));
  return 0.5f * x * ((1.f - t * t) * (0.79788456f + 0.1070322243f * x * x)) + 0.5f * (1.f + t);
}
__device__ __forceinline__ float sigmoid_f(float x) {
  return __builtin_amdgcn_rcpf(1.f + __expf(-x));
}

// ---------------------------------------------------------------------------
// Big GEMM: O(row,col) = sum_k A[row,k]*W[col,k]   (A: 8192x1024, W: 1024x1024)
// EPI==0: plain store. EPI==1: O = gelu(acc) * postLN(XQW[row,col]).
// Block tile 128x64, 8 waves, each wave: 16 rows x 64 cols (4 C-frags).
// Double-buffered k-tiles; async global->LDS (ASYNCcnt) when available.
// ---------------------------------------------------------------------------
template <int EPI>
__global__ __launch_bounds__(256) void gemm1024(
    const float* __restrict__ A, const float* __restrict__ W, float* __restrict__ O,
    const float* __restrict__ gateSrc, const float* __restrict__ rstats,
    const float* __restrict__ postw, const float* __restrict__ postb) {
  __shared__ float hA[2][128][32];
  __shared__ float wB[2][64][32];
  const int tid = threadIdx.x;
  const int lane = tid & 31, wave = tid >> 5;
  const int m = lane & 15, hi = lane >> 4, k0 = hi * 2;
  const int rowBase = blockIdx.x * 128;
  const int colBase = blockIdx.y * 64;
  v8f acc[4];
#pragma unroll
  for (int nt = 0; nt < 4; ++nt)
#pragma unroll
    for (int v = 0; v < 8; ++v) acc[nt][v] = 0.f;

#if USE_ASYNC_LDS
  auto stage_async = [&](int buf, int kb) {
#pragma unroll
    for (int i = 0; i < 4; ++i) {  // hA: 1024 x 16B chunks
      int e = tid + i * 256, r = e >> 3, c = (e & 7) * 4;
      __builtin_amdgcn_global_load_async_to_lds_b128(
          (gas_v4i_ptr)(A + (size_t)(rowBase + r) * WIDTH + kb + c),
          (lds_v4i_ptr)&hA[buf][r][c], 0, 0);
    }
#pragma unroll
    for (int i = 0; i < 2; ++i) {  // wB: 512 x 16B chunks
      int e = tid + i * 256, r = e >> 3, c = (e & 7) * 4;
      __builtin_amdgcn_global_load_async_to_lds_b128(
          (gas_v4i_ptr)(W + (size_t)(colBase + r) * WIDTH + kb + c),
          (lds_v4i_ptr)&wB[buf][r][c], 0, 0);
    }
  };
  stage_async(0, 0);
  __builtin_amdgcn_s_wait_asynccnt(0);
  __syncthreads();
#else
  {
    const float4* A4 = reinterpret_cast<const float4*>(A);
    const float4* W4 = reinterpret_cast<const float4*>(W);
#pragma unroll
    for (int i = 0; i < 4; ++i) {
      int e = tid + i * 256, r = e >> 3, c = (e & 7) * 4;
      *reinterpret_cast<float4*>(&hA[0][r][c]) = A4[(size_t)(rowBase + r) * (WIDTH / 4) + (c >> 2)];
    }
#pragma unroll
    for (int i = 0; i < 2; ++i) {
      int e = tid + i * 256, r = e >> 3, c = (e & 7) * 4;
      *reinterpret_cast<float4*>(&wB[0][r][c]) = W4[(size_t)(colBase + r) * (WIDTH / 4) + (c >> 2)];
    }
  }
  __syncthreads();
#endif

  for (int kb = 0; kb < WIDTH; kb += 32) {
    const int cur = (kb >> 5) & 1;
    const int nxt = cur ^ 1;
    const bool more = (kb + 32) < WIDTH;
#if USE_ASYNC_LDS
    if (more) stage_async(nxt, kb + 32);  // async engine fills next tile during compute
#else
    float4 ra[4], rw[2];
    if (more) {
      const float4* A4 = reinterpret_cast<const float4*>(A);
      const float4* W4 = reinterpret_cast<const float4*>(W);
#pragma unroll
      for (int i = 0; i < 4; ++i) {
        int e = tid + i * 256, r = e >> 3, c = (e & 7);
        ra[i] = A4[(size_t)(rowBase + r) * (WIDTH / 4) + ((kb + 32) >> 2) + c];
      }
#pragma unroll
      for (int i = 0; i < 2; ++i) {
        int e = tid + i * 256, r = e >> 3, c = (e & 7);
        rw[i] = W4[(size_t)(colBase + r) * (WIDTH / 4) + ((kb + 32) >> 2) + c];
      }
    }
#endif
#pragma unroll
    for (int kk = 0; kk < 8; ++kk) {
      v2f a = {hA[cur][wave * 16 + m][kk * 4 + k0], hA[cur][wave * 16 + m][kk * 4 + k0 + 1]};
#pragma unroll
      for (int nt = 0; nt < 4; ++nt) {
        v2f b = {wB[cur][nt * 16 + m][kk * 4 + k0], wB[cur][nt * 16 + m][kk * 4 + k0 + 1]};
        acc[nt] = wmma4(a, b, acc[nt]);
      }
    }
#if USE_ASYNC_LDS
    if (more) __builtin_amdgcn_s_wait_asynccnt(0);
#else
    if (more) {
#pragma unroll
      for (int i = 0; i < 4; ++i) {
        int e = tid + i * 256, r = e >> 3, c = (e & 7) * 4;
        *reinterpret_cast<float4*>(&hA[nxt][r][c]) = ra[i];
      }
#pragma unroll
      for (int i = 0; i < 2; ++i) {
        int e = tid + i * 256, r = e >> 3, c = (e & 7) * 4;
        *reinterpret_cast<float4*>(&wB[nxt][r][c]) = rw[i];
      }
    }
#endif
    __syncthreads();
  }

#pragma unroll
  for (int nt = 0; nt < 4; ++nt)
#pragma unroll
    for (int v = 0; v < 8; ++v) {
      int row = rowBase + wave * 16 + v + hi * 8;
      int col = colBase + nt * 16 + m;
      float x = acc[nt][v];
      if (EPI == 0) {
        O[(size_t)row * WIDTH + col] = x;
      } else {
        float mu = rstats[row * 2 + 0];
        float rs = rstats[row * 2 + 1];
        float q = gateSrc[(size_t)row * WIDTH + col];
        float ln = postw[col] * ((q - mu) * rs) + postb[col];
        O[(size_t)row * WIDTH + col] = gelu_f(x) * ln;
      }
    }
}

// ---------------------------------------------------------------------------
// per-token learning rate: LRE[(b*NH+h)*L + l] = BASE_LR*sigmoid(X.lr_w[h]+lr_b[h])/HD
// ---------------------------------------------------------------------------
__global__ __launch_bounds__(256) void lr_kernel(const float* __restrict__ H,
                                                 const float* __restrict__ lrw,
                                                 const float* __restrict__ lrb,
                                                 float* __restrict__ LRE) {
  int row = blockIdx.x;  // 0..B*L-1
  int b = row / LB, l = row % LB;
  int lane = threadIdx.x & 31, wave = threadIdx.x >> 5;
  const float* x = H + (size_t)row * WIDTH;
#pragma unroll
  for (int p = 0; p < 2; ++p) {
    int h = wave + p * 8;
    const float* w = lrw + (size_t)h * WIDTH;
    float s = 0.f;
    for (int c = lane; c < WIDTH; c += 32) s += x[c] * w[c];
#pragma unroll
    for (int off = 16; off > 0; off >>= 1) s += __shfl_down(s, off, 32);
    if (lane == 0) {
      float v = BASE_LRF * sigmoid_f(s + lrb[h]) / (float)HD;
      LRE[((size_t)b * NH + h) * LB + l] = v;
    }
  }
}

// ---------------------------------------------------------------------------
// row mean / rstd of XQW rows (for post layernorm). one wave per row.
// ---------------------------------------------------------------------------
__global__ __launch_bounds__(256) void rowstats_kernel(const float* __restrict__ X,
                                                       float* __restrict__ RST) {
  int lane = threadIdx.x & 31, wave = threadIdx.x >> 5;
  int row = blockIdx.x * 8 + wave;
  const float* x = X + (size_t)row * WIDTH;
  float s = 0.f, s2 = 0.f;
  for (int c = lane; c < WIDTH; c += 32) {
    float v = x[c];
    s += v;
    s2 += v * v;
  }
#pragma unroll
  for (int off = 16; off > 0; off >>= 1) {
    s += __shfl_down(s, off, 32);
    s2 += __shfl_down(s2, off, 32);
  }
  if (lane == 0) {
    float mu = s / (float)WIDTH;
    float var = s2 / (float)WIDTH - mu * mu;
    RST[row * 2 + 0] = mu;
    RST[row * 2 + 1] = rsqrtf(var + EPSF);
  }
}

// ---------------------------------------------------------------------------
// TTT scan. One block per (b,h). Fast-weight state W1(64x256), W2(256x64),
// b1, b2 live in LDS (~218KB, CDNA5 WGP has 320KB). 128 sequential chunks of
// 16 tokens; every matmul is 16x16x4 f32 WMMA tiled over 8 wave32s.
// ---------------------------------------------------------------------------
__global__ __launch_bounds__(256) void scan_kernel(
    const float* __restrict__ XQ, const float* __restrict__ XK, const float* __restrict__ XV,
    const float* __restrict__ LRE, const float* __restrict__ W1g, const float* __restrict__ b1g,
    const float* __restrict__ W2g, const float* __restrict__ b2g,
    const float* __restrict__ normw, const float* __restrict__ normb,
    const float* __restrict__ tokl, float* __restrict__ XQW) {
  __shared__ float W1s[HD][FF];  // 64KB  [k][n]
  __shared__ float W2s[FF][HD];  // 64KB  [k][n]
  __shared__ float b1s[FF];
  __shared__ float b2s[HD];
  __shared__ float lnwv[HD], lnbv[HD];
  __shared__ float Xq[MBS][HD], Xk[MBS][HD], Tg[MBS][HD];
  __shared__ float Z1[MBS][FF];
  __shared__ float X2[MBS][FF];
  __shared__ float gZ1[MBS][FF];
  __shared__ float X2b[MBS][FF];
  __shared__ float Z2[MBS][HD];  // also reused for Z2_bar
  __shared__ float gZ2[MBS][HD];
  __shared__ float A1[MBS][MBS];
  __shared__ float Rm[MBS][MBS];
  __shared__ float redA[MBS][MBS], redB[MBS][MBS];
  __shared__ float smu[MBS], srstd[MBS], sgxm[MBS], sgxxh[MBS];
  __shared__ float etat[MBS], lrej[MBS], lastv[MBS];

  const int bh = blockIdx.x;
  const int b = bh / NH, h = bh % NH;
  const int tid = threadIdx.x;
  const int lane = tid & 31, wave = tid >> 5;
  const int m = lane & 15, hi = lane >> 4, k0 = hi * 2;

  for (int e = tid; e < HD * FF; e += 256) W1s[e / FF][e % FF] = W1g[(size_t)h * HD * FF + e];
  for (int e = tid; e < FF * HD; e += 256) W2s[e / HD][e % HD] = W2g[(size_t)h * FF * HD + e];
  for (int e = tid; e < FF; e += 256) b1s[e] = b1g[(size_t)h * FF + e];
  if (tid < HD) {
    b2s[tid] = b2g[(size_t)h * HD + tid];
    lnwv[tid] = normw[(size_t)h * HD + tid];
    lnbv[tid] = normb[(size_t)h * HD + tid];
  }
  if (tid < MBS) {
    float v = 1.f / (float)(tid + 1) + tokl[tid];
    etat[tid] = fmaxf(v, 0.f);
  }
  __syncthreads();

  for (int n = 0; n < NCHUNK; ++n) {
    const int l0 = n * MBS;
    // -- load chunk ---------------------------------------------------------
    for (int e = tid; e < MBS * HD; e += 256) {
      int i = e >> 6, c = e & 63;
      size_t gi = ((size_t)b * LB + l0 + i) * WIDTH + h * HD + c;
      float xq = XQ[gi], xk = XK[gi], xv = XV[gi];
      Xq[i][c] = xq;
      Xk[i][c] = xk;
      Tg[i][c] = xv - xk;
    }
    if (tid < MBS) {
      float lre = LRE[((size_t)b * NH + h) * LB + l0 + tid];
      lrej[tid] = lre;
      lastv[tid] = etat[MBS - 1] * lre;
    }
    __syncthreads();

    // -- Z1 = Xk@W1 + b1 ; X2 = gelu(Z1)   (M16 N256 K64) -------------------
#pragma unroll
    for (int t = 0; t < 2; ++t) {
      int nb = (wave * 2 + t) * 16;
      v8f acc;
      for (int v = 0; v < 8; ++v) acc[v] = 0.f;
#pragma unroll
      for (int kk = 0; kk < 16; ++kk) {
        v2f a = {Xk[m][kk * 4 + k0], Xk[m][kk * 4 + k0 + 1]};
        v2f bb = {W1s[kk * 4 + k0][nb + m], W1s[kk * 4 + k0 + 1][nb + m]};
        acc = wmma4(a, bb, acc);
      }
#pragma unroll
      for (int v = 0; v < 8; ++v) {
        int r = v + hi * 8, c = nb + m;
        float z = acc[v] + b1s[c];
        Z1[r][c] = z;
        X2[r][c] = gelu_f(z);
      }
    }
    __syncthreads();

    // -- Z2 = X2@W2 + b2   (M16 N64 K256), 4 waves --------------------------
    if (wave < 4) {
      int nb = wave * 16;
      v8f acc;
      for (int v = 0; v < 8; ++v) acc[v] = 0.f;
#pragma unroll
      for (int kk = 0; kk < 64; ++kk) {
        v2f a = {X2[m][kk * 4 + k0], X2[m][kk * 4 + k0 + 1]};
        v2f bb = {W2s[kk * 4 + k0][nb + m], W2s[kk * 4 + k0 + 1][nb + m]};
        acc = wmma4(a, bb, acc);
      }
#pragma unroll
      for (int v = 0; v < 8; ++v) Z2[v + hi * 8][nb + m] = acc[v] + b2s[nb + m];
    }
    __syncthreads();

    // -- gZ2 = ln_l2_bwd(Z2, Tg) --------------------------------------------
    {
      int r = tid >> 4, j = tid & 15;
      float s = 0.f, s2 = 0.f;
      for (int c = j * 4; c < j * 4 + 4; ++c) {
        float z = Z2[r][c];
        s += z;
        s2 += z * z;
      }
      redA[r][j] = s;
      redB[r][j] = s2;
      __syncthreads();
      if (tid < 16) {
        float a = 0.f, c2 = 0.f;
        for (int q = 0; q < 16; ++q) {
          a += redA[tid][q];
          c2 += redB[tid][q];
        }
        float mu = a / (float)HD;
        smu[tid] = mu;
        srstd[tid] = rsqrtf(c2 / (float)HD - mu * mu + EPSF);
      }
      __syncthreads();
      float mu = smu[r], rs = srstd[r];
      float gs = 0.f, gxs = 0.f;
      for (int c = j * 4; c < j * 4 + 4; ++c) {
        float xh = (Z2[r][c] - mu) * rs;
        float gx = (lnwv[c] * xh + lnbv[c] - Tg[r][c]) * lnwv[c];
        gZ2[r][c] = gx;
        gs += gx;
        gxs += gx * xh;
      }
      redA[r][j] = gs;
      redB[r][j] = gxs;
      __syncthreads();
      if (tid < 16) {
        float a = 0.f, c2 = 0.f;
        for (int q = 0; q < 16; ++q) {
          a += redA[tid][q];
          c2 += redB[tid][q];
        }
        sgxm[tid] = a / (float)HD;
        sgxxh[tid] = c2 / (float)HD;
      }
      __syncthreads();
      float gm = sgxm[r], gxh = sgxxh[r];
      for (int c = j * 4; c < j * 4 + 4; ++c) {
        float xh = (Z2[r][c] - mu) * rs;
        gZ2[r][c] = (gZ2[r][c] - gm - xh * gxh) * rs;
      }
    }
    __syncthreads();

    // -- gZ1 = (gZ2 @ W2^T) * gelu'(Z1)   (M16 N256 K64) --------------------
#pragma unroll
    for (int t = 0; t < 2; ++t) {
      int nb = (wave * 2 + t) * 16;
      v8f acc;
      for (int v = 0; v < 8; ++v) acc[v] = 0.f;
#pragma unroll
      for (int kk = 0; kk < 16; ++kk) {
        v2f a = {gZ2[m][kk * 4 + k0], gZ2[m][kk * 4 + k0 + 1]};
        v2f bb = {W2s[nb + m][kk * 4 + k0], W2s[nb + m][kk * 4 + k0 + 1]};  // B=W2^T
        acc = wmma4(a, bb, acc);
      }
#pragma unroll
      for (int v = 0; v < 8; ++v) {
        int r = v + hi * 8, c = nb + m;
        gZ1[r][c] = acc[v] * gelu_bwd_f(Z1[r][c]);
      }
    }
    __syncthreads();

    // -- Attn1 = Xq @ Xk^T (raw), wave 0 ------------------------------------
    if (wave == 0) {
      v8f acc;
      for (int v = 0; v < 8; ++v) acc[v] = 0.f;
#pragma unroll
      for (int kk = 0; kk < 16; ++kk) {
        v2f a = {Xq[m][kk * 4 + k0], Xq[m][kk * 4 + k0 + 1]};
        v2f bb = {Xk[m][kk * 4 + k0], Xk[m][kk * 4 + k0 + 1]};  // B=Xk^T
        acc = wmma4(a, bb, acc);
      }
#pragma unroll
      for (int v = 0; v < 8; ++v) A1[v + hi * 8][m] = acc[v];
    }
    __syncthreads();
    {  // Rm = -eta_tril * (Attn1 + 1)
      int i = tid >> 4, j = tid & 15;
      float e = (j <= i) ? etat[i] * lrej[j] : 0.f;
      Rm[i][j] = -e * (A1[i][j] + 1.f);
    }
    __syncthreads();

    // -- X2b = gelu(Xq@W1 + Rm@gZ1 + b1)   (M16 N256 K64+16) ----------------
#pragma unroll
    for (int t = 0; t < 2; ++t) {
      int nb = (wave * 2 + t) * 16;
      v8f acc;
      for (int v = 0; v < 8; ++v) acc[v] = 0.f;
#pragma unroll
      for (int kk = 0; kk < 16; ++kk) {
        v2f a = {Xq[m][kk * 4 + k0], Xq[m][kk * 4 + k0 + 1]};
        v2f bb = {W1s[kk * 4 + k0][nb + m], W1s[kk * 4 + k0 + 1][nb + m]};
        acc = wmma4(a, bb, acc);
      }
#pragma unroll
      for (int kk = 0; kk < 4; ++kk) {
        v2f a = {Rm[m][kk * 4 + k0], Rm[m][kk * 4 + k0 + 1]};
        v2f bb = {gZ1[kk * 4 + k0][nb + m], gZ1[kk * 4 + k0 + 1][nb + m]};
        acc = wmma4(a, bb, acc);
      }
#pragma unroll
      for (int v = 0; v < 8; ++v) {
        int r = v + hi * 8, c = nb + m;
        X2b[r][c] = gelu_f(acc[v] + b1s[c]);
      }
    }
    __syncthreads();

    // -- Attn2 = X2b @ X2^T (raw), wave 0 -----------------------------------
    if (wave == 0) {
      v8f acc;
      for (int v = 0; v < 8; ++v) acc[v] = 0.f;
#pragma unroll
      for (int kk = 0; kk < 64; ++kk) {
        v2f a = {X2b[m][kk * 4 + k0], X2b[m][kk * 4 + k0 + 1]};
        v2f bb = {X2[m][kk * 4 + k0], X2[m][kk * 4 + k0 + 1]};  // B=X2^T
        acc = wmma4(a, bb, acc);
      }
#pragma unroll
      for (int v = 0; v < 8; ++v) A1[v + hi * 8][m] = acc[v];
    }
    __syncthreads();
    {
      int i = tid >> 4, j = tid & 15;
      float e = (j <= i) ? etat[i] * lrej[j] : 0.f;
      Rm[i][j] = -e * (A1[i][j] + 1.f);
    }
    __syncthreads();

    // -- Z2_bar = X2b@W2 + Rm@gZ2 + b2   (M16 N64 K256+16), 4 waves ---------
    if (wave < 4) {
      int nb = wave * 16;
      v8f acc;
      for (int v = 0; v < 8; ++v) acc[v] = 0.f;
#pragma unroll
      for (int kk = 0; kk < 64; ++kk) {
        v2f a = {X2b[m][kk * 4 + k0], X2b[m][kk * 4 + k0 + 1]};
        v2f bb = {W2s[kk * 4 + k0][nb + m], W2s[kk * 4 + k0 + 1][nb + m]};
        acc = wmma4(a, bb, acc);
      }
#pragma unroll
      for (int kk = 0; kk < 4; ++kk) {
        v2f a = {Rm[m][kk * 4 + k0], Rm[m][kk * 4 + k0 + 1]};
        v2f bb = {gZ2[kk * 4 + k0][nb + m], gZ2[kk * 4 + k0 + 1][nb + m]};
        acc = wmma4(a, bb, acc);
      }
#pragma unroll
      for (int v = 0; v < 8; ++v) Z2[v + hi * 8][nb + m] = acc[v] + b2s[nb + m];
    }
    __syncthreads();

    // -- XQW chunk = Xq + ln_fwd(Z2_bar) ------------------------------------
    {
      int r = tid >> 4, j = tid & 15;
      float s = 0.f, s2 = 0.f;
      for (int c = j * 4; c < j * 4 + 4; ++c) {
        float z = Z2[r][c];
        s += z;
        s2 += z * z;
      }
      redA[r][j] = s;
      redB[r][j] = s2;
      __syncthreads();
      if (tid < 16) {
        float a = 0.f, c2 = 0.f;
        for (int q = 0; q < 16; ++q) {
          a += redA[tid][q];
          c2 += redB[tid][q];
        }
        float mu = a / (float)HD;
        smu[tid] = mu;
        srstd[tid] = rsqrtf(c2 / (float)HD - mu * mu + EPSF);
      }
      __syncthreads();
      float mu = smu[r], rs = srstd[r];
      for (int c = j * 4; c < j * 4 + 4; ++c) {
        float xh = (Z2[r][c] - mu) * rs;
        XQW[((size_t)b * LB + l0 + r) * WIDTH + h * HD + c] = Xq[r][c] + lnwv[c] * xh + lnbv[c];
      }
    }
    __syncthreads();

    // -- state updates: W1 -= (last*Xk)^T@gZ1, W2 -= (last*X2)^T@gZ2 --------
#pragma unroll
    for (int jx = 0; jx < 8; ++jx) {  // W1: 4x16 tile jobs over 8 waves
      int job = wave * 8 + jx;
      int mt = job >> 4, nt = job & 15;
      int mb = mt * 16, nb = nt * 16;
      v8f acc;
#pragma unroll
      for (int v = 0; v < 8; ++v) acc[v] = W1s[mb + v + hi * 8][nb + m];
#pragma unroll
      for (int kk = 0; kk < 4; ++kk) {
        int ka = kk * 4 + k0;
        v2f a = {-lastv[ka] * Xk[ka][mb + m], -lastv[ka + 1] * Xk[ka + 1][mb + m]};
        v2f bb = {gZ1[ka][nb + m], gZ1[ka + 1][nb + m]};
        acc = wmma4(a, bb, acc);
      }
#pragma unroll
      for (int v = 0; v < 8; ++v) W1s[mb + v + hi * 8][nb + m] = acc[v];
    }
#pragma unroll
    for (int jx = 0; jx < 8; ++jx) {  // W2: 16x4 tiles
      int job = wave * 8 + jx;
      int mt = job >> 2, nt = job & 3;
      int mb = mt * 16, nb = nt * 16;
      v8f acc;
#pragma unroll
      for (int v = 0; v < 8; ++v) acc[v] = W2s[mb + v + hi * 8][nb + m];
#pragma unroll
      for (int kk = 0; kk < 4; ++kk) {
        int ka = kk * 4 + k0;
        v2f a = {-lastv[ka] * X2[ka][mb + m], -lastv[ka + 1] * X2[ka + 1][mb + m]};
        v2f bb = {gZ2[ka][nb + m], gZ2[ka + 1][nb + m]};
        acc = wmma4(a, bb, acc);
      }
#pragma unroll
      for (int v = 0; v < 8; ++v) W2s[mb + v + hi * 8][nb + m] = acc[v];
    }
    {  // bias updates
      float s = 0.f;
#pragma unroll
      for (int i = 0; i < MBS; ++i) s += lastv[i] * gZ1[i][tid];
      b1s[tid] -= s;
      if (tid < HD) {
        float s2 = 0.f;
#pragma unroll
        for (int i = 0; i < MBS; ++i) s2 += lastv[i] * gZ2[i][tid];
        b2s[tid] -= s2;
      }
    }
    __syncthreads();
  }
}

// ---------------------------------------------------------------------------
extern "C" void kernel_launch(void* const* d_in, const int* in_sizes, int n_in,
                              void* d_out, int out_size, void* d_ws, size_t ws_size,
                              hipStream_t stream) {
  const float* H     = (const float*)d_in[0];
  const float* Wq    = (const float*)d_in[1];
  const float* Wk    = (const float*)d_in[2];
  const float* Wv    = (const float*)d_in[3];
  const float* Wo    = (const float*)d_in[4];
  const float* Wg    = (const float*)d_in[5];
  const float* lrw   = (const float*)d_in[6];
  const float* lrb   = (const float*)d_in[7];
  const float* tokl  = (const float*)d_in[8];
  const float* normw = (const float*)d_in[9];
  const float* normb = (const float*)d_in[10];
  const float* W1    = (const float*)d_in[11];
  const float* b1    = (const float*)d_in[12];
  const float* W2    = (const float*)d_in[13];
  const float* b2    = (const float*)d_in[14];
  const float* postw = (const float*)d_in[15];
  const float* postb = (const float*)d_in[16];
  float* out = (float*)d_out;

  float* ws = (float*)d_ws;
  const size_t P = (size_t)BB * LB * WIDTH;  // 8M floats
  float* XQ   = ws;
  float* XK   = ws + P;
  float* XV   = ws + 2 * P;
  float* XQW  = ws + 3 * P;
  float* Tbuf = ws + 4 * P;
  float* LRE  = ws + 5 * P;                  // B*NH*L floats
  float* RST  = LRE + (size_t)BB * NH * LB;  // B*L*2 floats

  dim3 gg((BB * LB) / 128, WIDTH / 64, 1);

  gemm1024<0><<<gg, 256, 0, stream>>>(H, Wq, XQ, nullptr, nullptr, nullptr, nullptr);
  gemm1024<0><<<gg, 256, 0, stream>>>(H, Wk, XK, nullptr, nullptr, nullptr, nullptr);
  gemm1024<0><<<gg, 256, 0, stream>>>(H, Wv, XV, nullptr, nullptr, nullptr, nullptr);
  lr_kernel<<<BB * LB, 256, 0, stream>>>(H, lrw, lrb, LRE);
  scan_kernel<<<BB * NH, 256, 0, stream>>>(XQ, XK, XV, LRE, W1, b1, W2, b2, normw, normb,
                                           tokl, XQW);
  rowstats_kernel<<<(BB * LB) / 8, 256, 0, stream>>>(XQW, RST);
  gemm1024<1><<<gg, 256, 0, stream>>>(H, Wg, Tbuf, XQW, RST, postw, postb);
  gemm1024<0><<<gg, 256, 0, stream>>>(Tbuf, Wo, out, nullptr, nullptr, nullptr, nullptr);
}